// UpsamplingLayer_64931315581435
// MI455X (gfx1250) — compile-verified
//
#include <hip/hip_runtime.h>
#include <math.h>

// ---------------------------------------------------------------------------
// 3-NN inverse-distance flow upsampling for MI455X (gfx1250, wave32).
//
// Selection key per (sparse m, dense n) computed entirely inside one
// V_WMMA_F32_16X16X4_F32:   key = |s_m|^2 - 2 * <s_m, x_n>
//   A (16x4, M=sparse): row m = ( sx, sy, sz, |s|^2 )
//   B (4x16, N=dense) : col n = (-2x, -2y, -2z, 1 )
// key ordering == d^2 ordering (dense |x|^2 is constant per column).
//
// Inner loop is software-pipelined (A operands prefetched one iteration
// ahead) and processes two sparse tiles per trip with independent
// accumulators so the ds_load latency and the WMMA->VALU hazard are both
// covered by useful work.
// ---------------------------------------------------------------------------

typedef __attribute__((ext_vector_type(2))) float v2f;
typedef __attribute__((ext_vector_type(8))) float v8f;

#define BATCH            2
#define NDENSE           16384
#define NSPARSE          4096
#define TILES            (NSPARSE / 16)          // 256 WMMA tiles per wave
#define WAVES_PER_BLOCK  8
#define DENSE_PER_BLOCK  (WAVES_PER_BLOCK * 16)  // 128 dense points / block
#define BLOCKS_PER_BATCH (NDENSE / DENSE_PER_BLOCK)

__device__ __forceinline__ void insert3(float c, int idx,
                                        float& d0, float& d1, float& d2,
                                        int& i0, int& i1, int& i2) {
  // keep (d0<=d1<=d2); strict < keeps earlier (lower) index on ties,
  // matching jax.lax.top_k tie-breaking.
  const bool l2 = c < d2, l1 = c < d1, l0 = c < d0;
  d2 = l2 ? (l1 ? d1 : c) : d2;   i2 = l2 ? (l1 ? i1 : idx) : i2;
  d1 = l1 ? (l0 ? d0 : c) : d1;   i1 = l1 ? (l0 ? i0 : idx) : i1;
  d0 = l0 ? c : d0;               i0 = l0 ? idx : i0;
}

__global__ __launch_bounds__(256) void upsample_flow_3nn_kernel(
    const float* __restrict__ xyz,     // [B,3,N]
    const float* __restrict__ sxyz,    // [B,3,S]
    const float* __restrict__ sflow,   // [B,3,S]
    float* __restrict__ out) {         // [B,3,N]
  // Sparse set staged in LDS as float4 {x, y, z, |s|^2}: 64 KB (of 320 KB/WGP).
  __shared__ float4 s4[NSPARSE];

  const int b    = blockIdx.x / BLOCKS_PER_BATCH;
  const int blk  = blockIdx.x % BLOCKS_PER_BATCH;
  const int tid  = threadIdx.x;
  const int lane = tid & 31;
  const int wave = tid >> 5;
  const int p    = lane & 15;   // point-in-tile (both matrix halves)
  const int hi   = lane >> 4;   // 0: K=0,1 half   1: K=2,3 half

  // ---- stage sparse coords + norms into LDS (coalesced per coord plane) ----
  const float* sx = sxyz + b * 3 * NSPARSE;
  for (int s = tid; s < NSPARSE; s += 256) {
    const float x = sx[s];
    const float y = sx[NSPARSE + s];
    const float z = sx[2 * NSPARSE + s];
    s4[s] = make_float4(x, y, z, fmaf(x, x, fmaf(y, y, z * z)));
  }
  __syncthreads();

  // ---- loop-invariant B operand: this wave's 16 dense points --------------
  const int n = blk * DENSE_PER_BLOCK + wave * 16 + p;  // dense point (lane & partner)
  const float* dxp = xyz + b * 3 * NDENSE;
  const float xd = dxp[n];
  const float yd = dxp[NDENSE + n];
  const float zd = dxp[2 * NDENSE + n];

  v2f bm;
  bm.x = hi ? (-2.0f * zd) : (-2.0f * xd);   // K=2 : K=0
  bm.y = hi ? 1.0f         : (-2.0f * yd);   // K=3 : K=1

  // ---- streaming top-3 over all sparse tiles ------------------------------
  float d0 = INFINITY, d1 = INFINITY, d2 = INFINITY;
  int   i0 = 0, i1 = 0, i2 = 0;

  // Per-lane A-operand stream: tile t lives at aptr + t*64 floats.
  const float* aptr = (const float*)s4 + (p << 2) + (hi << 1);

  float2 a0 = *(const float2*)(aptr);        // tile 0
  float2 a1 = *(const float2*)(aptr + 64);   // tile 1

  for (int t = 0; t < TILES; t += 2) {
    // Prefetch the next pair (wraps in-bounds on the final trip).
    const int tn = (t + 2) & (TILES - 1);
    const float2 nx0 = *(const float2*)(aptr + tn * 64);
    const float2 nx1 = *(const float2*)(aptr + tn * 64 + 64);

    v2f am0; am0.x = a0.x; am0.y = a0.y;
    v2f am1; am1.x = a1.x; am1.y = a1.y;

    v8f acc0 = {0.f, 0.f, 0.f, 0.f, 0.f, 0.f, 0.f, 0.f};
    v8f acc1 = {0.f, 0.f, 0.f, 0.f, 0.f, 0.f, 0.f, 0.f};
    acc0 = __builtin_amdgcn_wmma_f32_16x16x4_f32(false, am0, false, bm,
                                                 (short)0, acc0, false, false);
    acc1 = __builtin_amdgcn_wmma_f32_16x16x4_f32(false, am1, false, bm,
                                                 (short)0, acc1, false, false);

    // C layout: VGPR r -> sparse row (r + 8*hi) of its tile, column = p.
    const float m0 = fminf(fminf(fminf(acc0[0], acc0[1]), fminf(acc0[2], acc0[3])),
                           fminf(fminf(acc0[4], acc0[5]), fminf(acc0[6], acc0[7])));
    const float m1 = fminf(fminf(fminf(acc1[0], acc1[1]), fminf(acc1[2], acc1[3])),
                           fminf(fminf(acc1[4], acc1[5]), fminf(acc1[6], acc1[7])));
    if (__any(fminf(m0, m1) < d2)) {   // uniform wave branch; rare after warmup
      const int base0 = t * 16 + hi * 8;
      insert3(acc0[0], base0 + 0, d0, d1, d2, i0, i1, i2);
      insert3(acc0[1], base0 + 1, d0, d1, d2, i0, i1, i2);
      insert3(acc0[2], base0 + 2, d0, d1, d2, i0, i1, i2);
      insert3(acc0[3], base0 + 3, d0, d1, d2, i0, i1, i2);
      insert3(acc0[4], base0 + 4, d0, d1, d2, i0, i1, i2);
      insert3(acc0[5], base0 + 5, d0, d1, d2, i0, i1, i2);
      insert3(acc0[6], base0 + 6, d0, d1, d2, i0, i1, i2);
      insert3(acc0[7], base0 + 7, d0, d1, d2, i0, i1, i2);
      const int base1 = base0 + 16;
      insert3(acc1[0], base1 + 0, d0, d1, d2, i0, i1, i2);
      insert3(acc1[1], base1 + 1, d0, d1, d2, i0, i1, i2);
      insert3(acc1[2], base1 + 2, d0, d1, d2, i0, i1, i2);
      insert3(acc1[3], base1 + 3, d0, d1, d2, i0, i1, i2);
      insert3(acc1[4], base1 + 4, d0, d1, d2, i0, i1, i2);
      insert3(acc1[5], base1 + 5, d0, d1, d2, i0, i1, i2);
      insert3(acc1[6], base1 + 6, d0, d1, d2, i0, i1, i2);
      insert3(acc1[7], base1 + 7, d0, d1, d2, i0, i1, i2);
    }
    a0 = nx0;
    a1 = nx1;
  }

  // ---- merge the two half-columns (lane n holds rows 0-7, lane n+16: 8-15)
  const float pd0 = __shfl_xor(d0, 16, 32);
  const float pd1 = __shfl_xor(d1, 16, 32);
  const float pd2 = __shfl_xor(d2, 16, 32);
  const int   pi0 = __shfl_xor(i0, 16, 32);
  const int   pi1 = __shfl_xor(i1, 16, 32);
  const int   pi2 = __shfl_xor(i2, 16, 32);
  insert3(pd0, pi0, d0, d1, d2, i0, i1, i2);
  insert3(pd1, pi1, d0, d1, d2, i0, i1, i2);
  insert3(pd2, pi2, d0, d1, d2, i0, i1, i2);

  // ---- final gather + inverse-distance weights (lanes 0-15 only) ----------
  if (lane < 16) {
    const float4 s0 = s4[i0];
    const float4 s1 = s4[i1];
    const float4 s2 = s4[i2];

    float ax = s0.x - xd, ay = s0.y - yd, az = s0.z - zd;
    float r0 = sqrtf(fmaf(ax, ax, fmaf(ay, ay, az * az)));
    ax = s1.x - xd; ay = s1.y - yd; az = s1.z - zd;
    float r1 = sqrtf(fmaf(ax, ax, fmaf(ay, ay, az * az)));
    ax = s2.x - xd; ay = s2.y - yd; az = s2.z - zd;
    float r2 = sqrtf(fmaf(ax, ax, fmaf(ay, ay, az * az)));

    r0 = fmaxf(r0, 1e-10f); r1 = fmaxf(r1, 1e-10f); r2 = fmaxf(r2, 1e-10f);
    const float w0 = 1.0f / r0, w1 = 1.0f / r1, w2 = 1.0f / r2;
    const float inv_sum = 1.0f / (w0 + w1 + w2);

    const float* fp = sflow + b * 3 * NSPARSE;
    const float fx = (w0 * fp[i0] + w1 * fp[i1] + w2 * fp[i2]) * inv_sum;
    const float fy = (w0 * fp[NSPARSE + i0] + w1 * fp[NSPARSE + i1] +
                      w2 * fp[NSPARSE + i2]) * inv_sum;
    const float fz = (w0 * fp[2 * NSPARSE + i0] + w1 * fp[2 * NSPARSE + i1] +
                      w2 * fp[2 * NSPARSE + i2]) * inv_sum;

    float* op = out + b * 3 * NDENSE;
    op[n]              = fx;
    op[NDENSE + n]     = fy;
    op[2 * NDENSE + n] = fz;
  }
}

extern "C" void kernel_launch(void* const* d_in, const int* in_sizes, int n_in,
                              void* d_out, int out_size, void* d_ws, size_t ws_size,
                              hipStream_t stream) {
  (void)in_sizes; (void)n_in; (void)out_size; (void)d_ws; (void)ws_size;
  const float* xyz   = (const float*)d_in[0];  // [2,3,16384]
  const float* sxyz  = (const float*)d_in[1];  // [2,3,4096]
  const float* sflow = (const float*)d_in[2];  // [2,3,4096]
  float* out = (float*)d_out;                  // [2,3,16384]

  dim3 grid(BATCH * BLOCKS_PER_BATCH);         // 256 blocks
  dim3 block(256);                             // 8 wave32 / block
  upsample_flow_3nn_kernel<<<grid, block, 0, stream>>>(xyz, sxyz, sflow, out);
}